// ZConditionedLinear_52063593562966
// MI455X (gfx1250) — compile-verified
//
#include <hip/hip_runtime.h>

typedef __attribute__((ext_vector_type(2))) float        v2f;
typedef __attribute__((ext_vector_type(8))) float        v8f;
typedef __attribute__((ext_vector_type(4))) unsigned int v4u;
typedef __attribute__((ext_vector_type(8))) int          v8i;
typedef __attribute__((ext_vector_type(4))) int          v4i;

#define N_ATOMS_C   25000
#define IN_F        128
#define OUT_F       128
#define N_SPECIES   64
#define N_SLICES    8
#define SLICE_SZ    ((N_ATOMS_C + N_SLICES - 1) / N_SLICES)   // 3125
#define WPAD        132                                        // padded row stride (floats)
#define LIST_CAP    3136                                       // >= SLICE_SZ, multiple of 16

// dynamic LDS layout (floats):
//   sW    : OUT_F * WPAD = 16896 floats (67584 B)   <- filled by TDM with pad
//   sF    : 16 * WPAD    =  2112 floats ( 8448 B)
//   sList : LIST_CAP ints (12544 B)
//   sCount: 1 int (+pad)
#define SMEM_BYTES ((OUT_F * WPAD + 16 * WPAD) * 4 + LIST_CAP * 4 + 16)

__global__ __launch_bounds__(256, 1)
void zcond_linear_wmma(const float* __restrict__ features,
                       const int*   __restrict__ species_idx,
                       const float* __restrict__ weight,
                       const float* __restrict__ bias,
                       float*       __restrict__ out)
{
    extern __shared__ float smem[];
    float* sW     = smem;                          // [OUT_F][WPAD]
    float* sF     = sW + OUT_F * WPAD;             // [16][WPAD]
    int*   sList  = (int*)(sF + 16 * WPAD);        // [LIST_CAP]
    int*   sCount = (int*)(sList + LIST_CAP);

    const int s     = blockIdx.x;          // species
    const int slice = blockIdx.y;          // atom range slice
    const int tid   = threadIdx.x;
    const int lane  = tid & 31;
    const int wave  = tid >> 5;            // 0..7 -> output column block
    const int half  = lane >> 4;           // 0 or 1 (lane half)
    const int l16   = lane & 15;

    if (tid == 0) sCount[0] = 0;
    __syncthreads();

    // ---- 1. compact matching atoms of this slice into LDS list ----
    const int a0 = slice * SLICE_SZ;
    const int a1 = (a0 + SLICE_SZ < N_ATOMS_C) ? (a0 + SLICE_SZ) : N_ATOMS_C;
    for (int a = a0 + tid; a < a1; a += 256) {
        if (species_idx[a] == s) {
            int p = atomicAdd(sCount, 1);
            sList[p] = a;
        }
    }
    __syncthreads();
    const int count = sCount[0];
    if (count == 0) return;                       // block-uniform exit, EXEC stays full
    const int padded = (count + 15) & ~15;
    for (int p = count + tid; p < padded; p += 256) sList[p] = -1;

    // ---- 2. TDM: DMA W[s] (128x128 f32) into LDS with native +4-DWORD row pad ----
    // D# per CDNA5 ISA. Wave 0 issues; TENSORcnt is per-wave,
    // so wave 0 waits, everyone else syncs at the tile-loop barrier.
    const float* Wg = weight + (size_t)s * OUT_F * IN_F;
    if (tid < 32) {
        const unsigned long long gaddr   = (unsigned long long)(uintptr_t)Wg;
        const unsigned           lds_off = (unsigned)(uintptr_t)sW;  // wave-relative LDS byte offset

        v4u g0;
        g0[0] = 1u;                                   // count=1, is_restore=0, gather off
        g0[1] = lds_off;                              // lds_addr [63:32]
        g0[2] = (unsigned)(gaddr & 0xffffffffu);      // global_addr[31:0]
        g0[3] = (unsigned)((gaddr >> 32) & 0x01ffffffu) | (2u << 30);  // addr[56:32] | type=2

        v8i g1;
        g1[0] = (2 << 16)      // data_size = 4 bytes
              | (1 << 20)      // pad_enable
              | (6 << 22)      // pad_interval: every 128 DWORDs
              | (3 << 25);     // pad_amount: 4 DWORDs  -> row stride 132 floats
        g1[1] = (int)(128u << 16);   // tensor_dim0 = 128 (bits 79:48 low half)
        g1[2] = (int)(128u << 16);   // tensor_dim0 hi=0 | tensor_dim1 = 128
        g1[3] = (int)(128u << 16);   // tensor_dim1 hi=0 | tile_dim0 = 128
        g1[4] = 128;                 // tile_dim1 = 128, tile_dim2 = 0 (2-D)
        g1[5] = 128;                 // tensor_dim0_stride[31:0] = 128 elements
        g1[6] = 0;                   // stride hi | tensor_dim1_stride lo = 0
        g1[7] = 0;

        v4i g2 = {0, 0, 0, 0};             // unused dims (zero-filled group 2)
        v4i g3 = {0, 0, 0, 0};             // unused dims (zero-filled group 3)
        v8i g4 = {0, 0, 0, 0, 0, 0, 0, 0}; // extra group on 6-arg toolchain (zero-filled)

        __builtin_amdgcn_tensor_load_to_lds(g0, g1, g2, g3, g4, 0);
        __builtin_amdgcn_s_wait_tensorcnt(0);
    }

    // per-wave bias: output column obase + l16 (same for both lane halves)
    const int obase = wave * 16;
    const float bval = bias[s * OUT_F + obase + l16];

    const int ntiles = padded >> 4;
    for (int t = 0; t < ntiles; ++t) {
        __syncthreads();   // covers: list-pad + TDM weight fill (t==0), sF reuse (t>0)

        // ---- 3a. stage 16x128 feature tile into LDS (coalesced, zero-pad) ----
        {
            const int m = tid >> 4;               // tile row 0..15
            const int c = (tid & 15) * 8;         // 8 floats per thread
            const int atom = sList[t * 16 + m];
            float* dst = sF + m * WPAD + c;
            if (atom >= 0) {
                const float4* src = (const float4*)(features + (size_t)atom * IN_F + c);
                const float4 v0 = src[0], v1 = src[1];
                dst[0] = v0.x; dst[1] = v0.y; dst[2] = v0.z; dst[3] = v0.w;
                dst[4] = v1.x; dst[5] = v1.y; dst[6] = v1.z; dst[7] = v1.w;
            } else {
#pragma unroll
                for (int j = 0; j < 8; ++j) dst[j] = 0.0f;
            }
        }
        __syncthreads();

        // ---- 3b. 16x16 output block per wave: K-loop of V_WMMA_F32_16X16X4_F32 ----
        // A: lane l -> row M=l16, K pair (k0 + 2*half, +1)
        // B: lane l -> col N=l16, K pair (k0 + 2*half, +1); B[k][n] = W[obase+n][k]
        v8f acc = {};
        const float* fRow = sF + l16 * WPAD + 2 * half;
        const float* wRow = sW + (obase + l16) * WPAD + 2 * half;
#pragma unroll 8
        for (int k0 = 0; k0 < IN_F; k0 += 4) {
            const v2f A = *(const v2f*)(fRow + k0);
            const v2f B = *(const v2f*)(wRow + k0);
            acc = __builtin_amdgcn_wmma_f32_16x16x4_f32(
                false, A, false, B, (short)0, acc, false, false);
        }

        // ---- 3c. store: VGPR r holds row M = r + 8*half, col obase+l16 ----
#pragma unroll
        for (int r = 0; r < 8; ++r) {
            const int m = r + 8 * half;
            const int atom = sList[t * 16 + m];
            if (atom >= 0) {
                out[(size_t)atom * OUT_F + obase + l16] = acc[r] + bval;
            }
        }
    }
}

extern "C" void kernel_launch(void* const* d_in, const int* in_sizes, int n_in,
                              void* d_out, int out_size, void* d_ws, size_t ws_size,
                              hipStream_t stream) {
    (void)in_sizes; (void)n_in; (void)d_ws; (void)ws_size; (void)out_size;
    const float* features    = (const float*)d_in[0];
    const int*   species_idx = (const int*)d_in[1];
    const float* weight      = (const float*)d_in[2];
    const float* bias        = (const float*)d_in[3];
    float*       out         = (float*)d_out;

    dim3 grid(N_SPECIES, N_SLICES);
    dim3 block(256);
    zcond_linear_wmma<<<grid, block, SMEM_BYTES, stream>>>(
        features, species_idx, weight, bias, out);
}